// GatingNetwork_8546984919172
// MI455X (gfx1250) — compile-verified
//
#include <hip/hip_runtime.h>

// Shapes from the reference
#define S_DIM 1024
#define B_DIM 256
#define N_DIM 256
#define M_DIM 512
#define E_DIM 64
#define SM1   (S_DIM - 1)   // 1023 frames
#define TEMP_MIN 1e-4f

typedef float v2f __attribute__((ext_vector_type(2)));
typedef float v8f __attribute__((ext_vector_type(8)));

// ---------------------------------------------------------------------------
// One K=4 step of a 16x16 fp32 WMMA (V_WMMA_F32_16X16X4_F32), with generic
// strides so the same helper serves row-major / transposed / strided operands.
// A layout (16x4):  lane<16 -> M=lane, K={0,1}; lane>=16 -> M=lane-16, K={2,3}
// B layout (4x16):  lane<16 -> N=lane, K={0,1}; lane>=16 -> N=lane-16, K={2,3}
// A element (m,k) at A[m*rsA + k*csA]; B element (k,n) at Bm[k*rsB + n*csB].
// ---------------------------------------------------------------------------
__device__ __forceinline__ v8f wmma_step(const float* A, int rsA, int csA,
                                         const float* Bm, int rsB, int csB,
                                         v8f acc) {
  const int lane = threadIdx.x & 31;
  const int half = lane >> 4;
  const int idx  = lane & 15;
  v2f a, b;
  a.x = A[idx * rsA + (2 * half)     * csA];
  a.y = A[idx * rsA + (2 * half + 1) * csA];
  b.x = Bm[(2 * half)     * rsB + idx * csB];
  b.y = Bm[(2 * half + 1) * rsB + idx * csB];
  return __builtin_amdgcn_wmma_f32_16x16x4_f32(false, a, false, b,
                                               (short)0, acc, false, false);
}

// ---------------------------------------------------------------------------
// k0: z_obs = D(256x512) @ z(512x256); stored TRANSPOSED as zt[b][n] so the
// distance kernel reads contiguous rows. 256 tiles of 16x16, 4 waves/block.
// ---------------------------------------------------------------------------
__global__ void k0_zobs(const float* __restrict__ D, const float* __restrict__ z,
                        float* __restrict__ zt) {
  const int wave = threadIdx.x >> 5;
  const int tile = blockIdx.x * 4 + wave;         // 0..255
  const int n0 = (tile >> 4) * 16;
  const int b0 = (tile & 15) * 16;
  v8f acc = {};
  for (int k = 0; k < M_DIM; k += 4)
    acc = wmma_step(D + n0 * M_DIM + k, M_DIM, 1,   // A = D (row-major)
                    z + k * B_DIM + b0, B_DIM, 1,   // B = z (row-major)
                    acc);
  const int lane = threadIdx.x & 31;
  const int half = lane >> 4, idx = lane & 15;
#pragma unroll
  for (int r = 0; r < 8; ++r) {
    const int n = n0 + r + 8 * half;   // output row (M)
    const int b = b0 + idx;            // output col (N)
    zt[b * N_DIM + n] = acc[r];        // transposed store
  }
}

// ---------------------------------------------------------------------------
// k1: dist[b][s] = sum_n |ctx[s][b][n] - zt[b][n]|, s in [0, S-2].
// One wave per (s,b): 32 lanes x 8 floats = 256 contiguous floats (2x float4).
// zt (256 KB) stays L2-resident across all 1023 s values.
// ---------------------------------------------------------------------------
__global__ void k1_dist(const float* __restrict__ ctx, const float* __restrict__ zt,
                        float* __restrict__ dist) {
  const int s = blockIdx.x;                       // 0..1022
  const int b = blockIdx.y * 8 + (threadIdx.x >> 5);
  const int lane = threadIdx.x & 31;
  const float4* cp = (const float4*)(ctx + (size_t)s * (B_DIM * N_DIM) + b * N_DIM);
  const float4* zp = (const float4*)(zt + b * N_DIM);
  const float4 ca = cp[lane * 2 + 0], cb = cp[lane * 2 + 1];
  const float4 za = zp[lane * 2 + 0], zb = zp[lane * 2 + 1];
  float ssum = fabsf(ca.x - za.x) + fabsf(ca.y - za.y) +
               fabsf(ca.z - za.z) + fabsf(ca.w - za.w) +
               fabsf(cb.x - zb.x) + fabsf(cb.y - zb.y) +
               fabsf(cb.z - zb.z) + fabsf(cb.w - zb.w);
#pragma unroll
  for (int off = 16; off; off >>= 1) ssum += __shfl_xor(ssum, off, 32);
  if (lane == 0) dist[(size_t)b * SM1 + s] = ssum;
}

// ---------------------------------------------------------------------------
// k2: in-place softmax over s (1023 values) per column b: attn = softmax(-d/t1)
// Each thread keeps its <=4 values in registers across max/sum/store phases.
// ---------------------------------------------------------------------------
__global__ void k2_softmax(float* __restrict__ attn, const float* __restrict__ t1) {
  const int b = blockIdx.x;
  const int t = threadIdx.x;
  const float temp = fmaxf(fabsf(t1[0]), TEMP_MIN);
  const float inv = -1.0f / temp;
  float* col = attn + (size_t)b * SM1;
  __shared__ float red[256];

  float vals[4];
  int cnt = 0;
  for (int s = t; s < SM1; s += 256) vals[cnt++] = col[s] * inv;

  float m = -3.4e38f;
  for (int k = 0; k < cnt; ++k) m = fmaxf(m, vals[k]);
  red[t] = m; __syncthreads();
  for (int w = 128; w; w >>= 1) { if (t < w) red[t] = fmaxf(red[t], red[t + w]); __syncthreads(); }
  m = red[0]; __syncthreads();

  float sum = 0.f;
  for (int k = 0; k < cnt; ++k) { vals[k] = __expf(vals[k] - m); sum += vals[k]; }
  red[t] = sum; __syncthreads();
  for (int w = 128; w; w >>= 1) { if (t < w) red[t] += red[t + w]; __syncthreads(); }
  const float rinv = 1.0f / red[0];

  cnt = 0;
  for (int s = t; s < SM1; s += 256) col[s] = vals[cnt++] * rinv;
}

// ---------------------------------------------------------------------------
// k3: single streaming pass over context computing BOTH weighted sums:
//   c0[b][i] = sum_{s<S-1} attn[s,b] * ctx[s,b,i]
//   c1[b][i] = sum_{s>=1} attn[s-1,b] * ctx[s,b,i]
// Block per b: 1024 threads = (16 s-phases) x (64 lanes x float4 channels).
// Each context element is read exactly once, as global_load_b128.
// ---------------------------------------------------------------------------
__global__ __launch_bounds__(1024) void k3_wsum(const float* __restrict__ ctx,
                                                const float* __restrict__ attn,
                                                float* __restrict__ c0,
                                                float* __restrict__ c1) {
  const int b  = blockIdx.x;
  const int g  = threadIdx.x & 63;    // channel group -> floats [4g, 4g+3]
  const int j  = threadIdx.x >> 6;    // s-phase 0..15
  const float* ab = attn + (size_t)b * SM1;
  float4 a0 = {0.f, 0.f, 0.f, 0.f};
  float4 a1 = {0.f, 0.f, 0.f, 0.f};
  for (int it = 0; it < S_DIM / 16; ++it) {
    const int s = j + 16 * it;
    const float4 v = *(const float4*)(ctx + (size_t)s * (B_DIM * N_DIM) +
                                      b * N_DIM + 4 * g);
    if (s < SM1) {                    // false only for s = 1023
      const float w = ab[s];
      a0.x = fmaf(w, v.x, a0.x); a0.y = fmaf(w, v.y, a0.y);
      a0.z = fmaf(w, v.z, a0.z); a0.w = fmaf(w, v.w, a0.w);
    }
    if (s >= 1) {                     // false only for s = 0
      const float w = ab[s - 1];
      a1.x = fmaf(w, v.x, a1.x); a1.y = fmaf(w, v.y, a1.y);
      a1.z = fmaf(w, v.z, a1.z); a1.w = fmaf(w, v.w, a1.w);
    }
  }
  __shared__ float4 r0[1024];
  __shared__ float4 r1[1024];
  r0[threadIdx.x] = a0; r1[threadIdx.x] = a1;
  __syncthreads();
  if (threadIdx.x < 64) {             // fold the 16 s-phases
    float4 s0 = r0[threadIdx.x], s1 = r1[threadIdx.x];
    for (int p = 1; p < 16; ++p) {
      const float4 u0 = r0[p * 64 + threadIdx.x];
      const float4 u1 = r1[p * 64 + threadIdx.x];
      s0.x += u0.x; s0.y += u0.y; s0.z += u0.z; s0.w += u0.w;
      s1.x += u1.x; s1.y += u1.y; s1.z += u1.z; s1.w += u1.w;
    }
    *(float4*)(c0 + b * N_DIM + 4 * threadIdx.x) = s0;
    *(float4*)(c1 + b * N_DIM + 4 * threadIdx.x) = s1;
  }
}

// ---------------------------------------------------------------------------
// k4: conv collapsed by linearity + softmax-sums-to-1:
//   emb(256x256) = c0 @ w0^T + c1 @ w1^T + conv_b
// conv_w layout (O,I,2): w_tap^T[k][o] = conv_w[o*512 + k*2 + tap].
// ---------------------------------------------------------------------------
__global__ void k4_embed(const float* __restrict__ c0, const float* __restrict__ c1,
                         const float* __restrict__ convw, const float* __restrict__ convb,
                         float* __restrict__ emb) {
  const int wave = threadIdx.x >> 5;
  const int tile = blockIdx.x * 4 + wave;          // 0..255
  const int b0 = (tile >> 4) * 16;
  const int o0 = (tile & 15) * 16;
  v8f acc = {};
  for (int k = 0; k < N_DIM; k += 4) {
    acc = wmma_step(c0 + b0 * N_DIM + k, N_DIM, 1,
                    convw + o0 * (2 * N_DIM) + k * 2 + 0, 2, 2 * N_DIM, acc);
    acc = wmma_step(c1 + b0 * N_DIM + k, N_DIM, 1,
                    convw + o0 * (2 * N_DIM) + k * 2 + 1, 2, 2 * N_DIM, acc);
  }
  const int lane = threadIdx.x & 31;
  const int half = lane >> 4, idx = lane & 15;
  const float cb = convb[o0 + idx];
#pragma unroll
  for (int r = 0; r < 8; ++r)
    emb[(b0 + r + 8 * half) * N_DIM + o0 + idx] = acc[r] + cb;
}

// ---------------------------------------------------------------------------
// k5: fused MLP + softmax. Block handles 16 rows of B, all 64 experts.
//   h = relu([emb | z^T] @ W1^T)  (K = 256 + 512), kept in LDS
//   mlp = h @ W2^T                (K = 64)
//   out[e][b] = softmax_e(-mlp[b][e] / t2)
// ---------------------------------------------------------------------------
__global__ void k5_mlp(const float* __restrict__ emb, const float* __restrict__ z,
                       const float* __restrict__ W1, const float* __restrict__ W2,
                       const float* __restrict__ t2, float* __restrict__ out) {
  __shared__ float hs[16 * 64];
  __shared__ float ms[16 * 64];
  const int b0 = blockIdx.x * 16;
  const int wave = threadIdx.x >> 5;
  const int e0 = wave * 16;                        // this wave's 16 experts
  const int lane = threadIdx.x & 31;
  const int half = lane >> 4, idx = lane & 15;

  // h tile: A rows = batch, K = 768 (first 256 from emb, rest from z^T)
  v8f acc = {};
  for (int k = 0; k < N_DIM; k += 4)
    acc = wmma_step(emb + b0 * N_DIM + k, N_DIM, 1,
                    W1 + e0 * (N_DIM + M_DIM) + k, 1, N_DIM + M_DIM, acc);
  for (int k = 0; k < M_DIM; k += 4)
    acc = wmma_step(z + k * B_DIM + b0, 1, B_DIM,       // A(m,kl) = z[k+kl][b0+m]
                    W1 + e0 * (N_DIM + M_DIM) + N_DIM + k, 1, N_DIM + M_DIM, acc);
#pragma unroll
  for (int r = 0; r < 8; ++r)
    hs[(r + 8 * half) * 64 + e0 + idx] = fmaxf(acc[r], 0.f);
  __syncthreads();

  // mlp tile: K = 64 over LDS-resident h
  v8f acc2 = {};
  for (int k = 0; k < E_DIM; k += 4)
    acc2 = wmma_step(hs + k, 64, 1,
                     W2 + e0 * E_DIM + k, 1, E_DIM, acc2);
#pragma unroll
  for (int r = 0; r < 8; ++r)
    ms[(r + 8 * half) * 64 + e0 + idx] = acc2[r];
  __syncthreads();

  // softmax over the 64 experts for each of the 16 batch rows
  const int t = threadIdx.x;
  if (t < 16) {
    const float temp = fmaxf(fabsf(t2[0]), TEMP_MIN);
    const float inv = -1.0f / temp;
    float mx = -3.4e38f;
    for (int e = 0; e < E_DIM; ++e) mx = fmaxf(mx, ms[t * 64 + e] * inv);
    float sum = 0.f;
    for (int e = 0; e < E_DIM; ++e) sum += __expf(ms[t * 64 + e] * inv - mx);
    const float rs = 1.0f / sum;
    for (int e = 0; e < E_DIM; ++e)
      out[e * B_DIM + b0 + t] = __expf(ms[t * 64 + e] * inv - mx) * rs;
  }
}

// ---------------------------------------------------------------------------
extern "C" void kernel_launch(void* const* d_in, const int* in_sizes, int n_in,
                              void* d_out, int out_size, void* d_ws, size_t ws_size,
                              hipStream_t stream) {
  const float* ctx   = (const float*)d_in[0];  // (S,B,N)
  const float* z     = (const float*)d_in[1];  // (M,B)
  const float* D     = (const float*)d_in[2];  // (N,M)
  const float* convw = (const float*)d_in[3];  // (N,N,2)
  const float* convb = (const float*)d_in[4];  // (N,)
  const float* W1    = (const float*)d_in[5];  // (E, M+N)
  const float* W2    = (const float*)d_in[6];  // (E, E)
  const float* t1    = (const float*)d_in[7];
  const float* t2    = (const float*)d_in[8];
  float* out = (float*)d_out;                  // (E,B)

  // workspace carve-up (floats): zt | attn | c0 | c1 | emb  (~2 MB total)
  float* ws   = (float*)d_ws;
  float* zt   = ws;                               // B*N     = 65536
  float* attn = zt + B_DIM * N_DIM;               // B*(S-1) = 261888
  float* c0   = attn + (size_t)B_DIM * SM1;       // B*N
  float* c1   = c0 + B_DIM * N_DIM;               // B*N
  float* emb  = c1 + B_DIM * N_DIM;               // B*N

  k0_zobs   <<<64, 128, 0, stream>>>(D, z, zt);
  k1_dist   <<<dim3(SM1, B_DIM / 8), 256, 0, stream>>>(ctx, zt, attn);
  k2_softmax<<<B_DIM, 256, 0, stream>>>(attn, t1);
  k3_wsum   <<<B_DIM, 1024, 0, stream>>>(ctx, attn, c0, c1);
  k4_embed  <<<64, 128, 0, stream>>>(c0, c1, convw, convb, emb);
  k5_mlp    <<<B_DIM / 16, 128, 0, stream>>>(emb, z, W1, W2, t2, out);
}